// SecondOrderMF_44659069943958
// MI455X (gfx1250) — compile-verified
//
#include <hip/hip_runtime.h>

#ifndef __has_builtin
#define __has_builtin(x) 0
#endif

#if __has_builtin(__builtin_amdgcn_global_load_async_to_lds_b128) && __has_builtin(__builtin_amdgcn_s_wait_asynccnt)
#define USE_ASYNC 1
#else
#define USE_ASYNC 0
#endif

constexpr int T  = 128;
constexpr int T2 = T * T;        // 16384
constexpr int T3 = T2 * T;       // 2097152
constexpr int B  = 4;
constexpr int NW = 8;            // waves per block (256 threads, wave32)

// ---------- small helpers ----------
__device__ __forceinline__ float4 ld4(const float* p) {
    return *reinterpret_cast<const float4*>(p);
}
__device__ __forceinline__ float4 mul4(float4 a, float4 b) {
    float4 r; r.x = a.x*b.x; r.y = a.y*b.y; r.z = a.z*b.z; r.w = a.w*b.w; return r;
}
__device__ __forceinline__ void fmaS4(float4& acc, float s, float4 v) {
    acc.x += s*v.x; acc.y += s*v.y; acc.z += s*v.z; acc.w += s*v.w;
}
__device__ __forceinline__ void fma44(float4& acc, float4 s, float4 v) {
    acc.x += s.x*v.x; acc.y += s.y*v.y; acc.z += s.z*v.z; acc.w += s.w*v.w;
}
__device__ __forceinline__ float dot4(float4 a, float4 b) {
    return a.x*b.x + a.y*b.y + a.z*b.z + a.w*b.w;
}
__device__ __forceinline__ float waveRed(float v) {
    v += __shfl_xor(v, 16, 32);
    v += __shfl_xor(v,  8, 32);
    v += __shfl_xor(v,  4, 32);
    v += __shfl_xor(v,  2, 32);
    v += __shfl_xor(v,  1, 32);
    return v;
}
__device__ __forceinline__ void atomAddF(float* p, float v) {
    unsafeAtomicAdd(p, v);   // global_atomic_add_f32
}

#if USE_ASYNC
#define AS1 __attribute__((address_space(1)))
#define AS3 __attribute__((address_space(3)))
typedef int v4i __attribute__((ext_vector_type(4)));
typedef AS1 v4i* g_v4i_p;   // global 128-bit vector pointer
typedef AS3 v4i* l_v4i_p;   // LDS 128-bit vector pointer
__device__ __forceinline__ void async_row_b128(const float* g, float* l) {
    g_v4i_p gp = (g_v4i_p)(unsigned long long)(const void*)g;
    l_v4i_p lp = (l_v4i_p)(unsigned int)(unsigned long long)(void*)l;
    __builtin_amdgcn_global_load_async_to_lds_b128(gp, lp, 0, 0);
}
#endif

// ---------- kernel S: q* = sigmoid(cur*), out* = base* ----------
__global__ __launch_bounds__(256)
void k_sigmoid_init(const float* __restrict__ cS, const float* __restrict__ cH,
                    const float* __restrict__ cT,
                    const float* __restrict__ bS, const float* __restrict__ bH,
                    const float* __restrict__ bT,
                    float* __restrict__ qs, float* __restrict__ qh, float* __restrict__ qt,
                    float* __restrict__ oS, float* __restrict__ oH, float* __restrict__ oT) {
    int idx = blockIdx.x * blockDim.x + threadIdx.x;     // 0 .. B*T2-1
    float s = cS[idx], h = cH[idx], t = cT[idx];
    qs[idx] = 1.f / (1.f + expf(-s));
    qh[idx] = 1.f / (1.f + expf(-h));
    qt[idx] = 1.f / (1.f + expf(-t));
    oS[idx] = bS[idx];
    oH[idx] = bH[idx];
    oT[idx] = bT[idx];
}

// ---------- kernel P: workgroup = (b, j), loop over i ----------
// Terms: A1: span[j,k] += sum_i qh[i,j]*r_psh      A2: span[j,k] += sum_i qt[i,k]*r_pst
//        B4: pt[j,k]   += sum_i qs[i,j]*f_pst
//        A3: ph[i,j]   += sum_k qs[j,k]*r_psh      B2: span[i,j] += sum_k qt[j,k]*f_pst
__global__ __launch_bounds__(256)
void k_P(const float* __restrict__ qs, const float* __restrict__ qh, const float* __restrict__ qt,
         const float* __restrict__ rpsh, const float* __restrict__ rpst, const float* __restrict__ mr,
         const float* __restrict__ fpst, const float* __restrict__ mf,
         float* __restrict__ oS, float* __restrict__ oH, float* __restrict__ oT) {
    const int b = blockIdx.x >> 7, j = blockIdx.x & (T - 1);
    const int tid = threadIdx.x, wave = tid >> 5, lane = tid & 31;
    const int k4 = lane * 4;

    __shared__ float qh_col[T];
    __shared__ float qs_col[T];
    __shared__ float redbuf[NW][T];
#if USE_ASYNC
    __shared__ float stage[2][NW][5][T];   // 5 streams staged per wave, double buffered
#endif

    if (tid < T) {
        qh_col[tid] = qh[(b*T + tid)*T + j];
        qs_col[tid] = qs[(b*T + tid)*T + j];
    }
    __syncthreads();

    const float4 qsj4 = ld4(qs + (b*T + j)*T + k4);   // row qs[b,j,:]
    const float4 qtj4 = ld4(qt + (b*T + j)*T + k4);   // row qt[b,j,:]

    float4 accS  = {0.f, 0.f, 0.f, 0.f};
    float4 accPT = {0.f, 0.f, 0.f, 0.f};
    const int base_bj = b*T3 + j*T;

    auto compute = [&](int i, float4 r1, float4 r2, float4 f2) {
        const float qhi = qh_col[i];
        const float qsi = qs_col[i];
        const float4 qti4 = ld4(qt + (b*T + i)*T + k4);
        fmaS4(accS, qhi, r1);        // A1
        fma44(accS, qti4, r2);       // A2
        fmaS4(accPT, qsi, f2);       // B4
        float dA3 = waveRed(dot4(qsj4, r1));   // -> ph[b,i,j]
        float dB2 = waveRed(dot4(qtj4, f2));   // -> span[b,i,j]
        if (lane == 0) {
            atomAddF(oH + (b*T + i)*T + j, dA3);
            atomAddF(oS + (b*T + i)*T + j, dB2);
        }
    };

#if USE_ASYNC
    const float* srcs[5] = {rpsh, mr, rpst, fpst, mf};
    auto issue = [&](int buf, int i) {
        const int off = base_bj + i*T2 + k4;
        #pragma unroll
        for (int r = 0; r < 5; ++r)
            async_row_b128(srcs[r] + off, &stage[buf][wave][r][k4]);
    };
    auto consume = [&](int buf, int i) {
        float4 mr4 = ld4(&stage[buf][wave][1][k4]);
        float4 r1  = mul4(ld4(&stage[buf][wave][0][k4]), mr4);
        float4 r2  = mul4(ld4(&stage[buf][wave][2][k4]), mr4);
        float4 mf4 = ld4(&stage[buf][wave][4][k4]);
        float4 f2  = mul4(ld4(&stage[buf][wave][3][k4]), mf4);
        compute(i, r1, r2, f2);
    };

    issue(0, wave);
    int sbuf = 0;
    #pragma unroll 1
    for (int step = 0; step < 15; ++step) {
        issue(sbuf ^ 1, wave + (step + 1)*NW);
        __builtin_amdgcn_s_wait_asynccnt(5);   // oldest batch of 5 complete (in-order)
        consume(sbuf, wave + step*NW);
        sbuf ^= 1;
    }
    __builtin_amdgcn_s_wait_asynccnt(0);
    consume(sbuf, wave + 15*NW);
#else
    #pragma unroll 1
    for (int step = 0; step < 16; ++step) {
        const int i   = wave + step*NW;
        const int off = base_bj + i*T2 + k4;
        if (step + 1 < 16) {
            const int poff = off + NW*T2;
            __builtin_prefetch(rpsh + poff, 0, 1);
            __builtin_prefetch(mr   + poff, 0, 1);
            __builtin_prefetch(rpst + poff, 0, 1);
            __builtin_prefetch(fpst + poff, 0, 1);
            __builtin_prefetch(mf   + poff, 0, 1);
        }
        float4 mr4 = ld4(mr + off);
        float4 r1  = mul4(ld4(rpsh + off), mr4);
        float4 r2  = mul4(ld4(rpst + off), mr4);
        float4 mf4 = ld4(mf + off);
        float4 f2  = mul4(ld4(fpst + off), mf4);
        compute(i, r1, r2, f2);
    }
#endif

    // combine per-wave accumulators, one atomic add per element per workgroup
    __syncthreads();
    *reinterpret_cast<float4*>(&redbuf[wave][k4]) = accS;
    __syncthreads();
    if (tid < T) {
        float s = 0.f;
        #pragma unroll
        for (int w = 0; w < NW; ++w) s += redbuf[w][tid];
        atomAddF(oS + (b*T + j)*T + tid, s);     // span[b,j,k]
    }
    __syncthreads();
    *reinterpret_cast<float4*>(&redbuf[wave][k4]) = accPT;
    __syncthreads();
    if (tid < T) {
        float s = 0.f;
        #pragma unroll
        for (int w = 0; w < NW; ++w) s += redbuf[w][tid];
        atomAddF(oT + (b*T + j)*T + tid, s);     // pt[b,j,k]
    }
}

// ---------- kernel Q: workgroup = (b, i), loop over j ----------
// Terms: B3: ph[i,k] += sum_j qs[i,j]*f_psh     D1: ph[i,k] += sum_j qh[j,k]*c_h
//        A4: pt[i,k] += sum_j qs[j,k]*r_pst     D2: pt[i,k] += sum_j qt[j,k]*c_t
//        B1: span[i,j] += sum_k qh[i,k]*f_psh   C1: ph[i,j] += sum_k qh[i,k]*s_h
//        C2: pt[i,j] += sum_k qt[i,k]*s_t
__global__ __launch_bounds__(256)
void k_Q(const float* __restrict__ qs, const float* __restrict__ qh, const float* __restrict__ qt,
         const float* __restrict__ fpsh, const float* __restrict__ mf,
         const float* __restrict__ sh,   const float* __restrict__ st, const float* __restrict__ ms,
         const float* __restrict__ ch,   const float* __restrict__ ct, const float* __restrict__ mc,
         const float* __restrict__ rpst, const float* __restrict__ mr,
         float* __restrict__ oS, float* __restrict__ oH, float* __restrict__ oT) {
    const int b = blockIdx.x >> 7, i = blockIdx.x & (T - 1);
    const int tid = threadIdx.x, wave = tid >> 5, lane = tid & 31;
    const int k4 = lane * 4;

    __shared__ float qs_i[T];
    __shared__ float redbuf[NW][T];
    if (tid < T) qs_i[tid] = qs[(b*T + i)*T + tid];
    __syncthreads();

    const float4 qhi4 = ld4(qh + (b*T + i)*T + k4);
    const float4 qti4 = ld4(qt + (b*T + i)*T + k4);

    float4 accPH = {0.f, 0.f, 0.f, 0.f};
    float4 accPT = {0.f, 0.f, 0.f, 0.f};
    const int base_bi = b*T3 + i*T2;

    #pragma unroll 1
    for (int j = wave; j < T; j += NW) {
        const int off = base_bi + j*T + k4;
        if (j + NW < T) {
            const int poff = off + NW*T;
            __builtin_prefetch(fpsh + poff, 0, 1);
            __builtin_prefetch(mf   + poff, 0, 1);
            __builtin_prefetch(sh   + poff, 0, 1);
            __builtin_prefetch(st   + poff, 0, 1);
            __builtin_prefetch(ms   + poff, 0, 1);
            __builtin_prefetch(ch   + poff, 0, 1);
            __builtin_prefetch(ct   + poff, 0, 1);
            __builtin_prefetch(mc   + poff, 0, 1);
            __builtin_prefetch(rpst + poff, 0, 1);
        }
        float4 mf4 = ld4(mf + off);
        float4 f1  = mul4(ld4(fpsh + off), mf4);
        float4 ms4 = ld4(ms + off);
        float4 s1  = mul4(ld4(sh + off), ms4);
        float4 s2  = mul4(ld4(st + off), ms4);
        float4 mc4 = ld4(mc + off);
        float4 c1  = mul4(ld4(ch + off), mc4);
        float4 c2  = mul4(ld4(ct + off), mc4);
        float4 mr4 = ld4(mr + off);
        float4 r2  = mul4(ld4(rpst + off), mr4);

        float4 qhj4 = ld4(qh + (b*T + j)*T + k4);
        float4 qtj4 = ld4(qt + (b*T + j)*T + k4);
        float4 qsj4 = ld4(qs + (b*T + j)*T + k4);

        fmaS4(accPH, qs_i[j], f1);   // B3
        fma44(accPH, qhj4, c1);      // D1
        fma44(accPT, qsj4, r2);      // A4
        fma44(accPT, qtj4, c2);      // D2

        float dB1 = waveRed(dot4(qhi4, f1));
        float dC1 = waveRed(dot4(qhi4, s1));
        float dC2 = waveRed(dot4(qti4, s2));
        if (lane == 0) {
            atomAddF(oS + (b*T + i)*T + j, dB1);
            atomAddF(oH + (b*T + i)*T + j, dC1);
            atomAddF(oT + (b*T + i)*T + j, dC2);
        }
    }

    __syncthreads();
    *reinterpret_cast<float4*>(&redbuf[wave][k4]) = accPH;
    __syncthreads();
    if (tid < T) {
        float s = 0.f;
        #pragma unroll
        for (int w = 0; w < NW; ++w) s += redbuf[w][tid];
        atomAddF(oH + (b*T + i)*T + tid, s);     // ph[b,i,k]
    }
    __syncthreads();
    *reinterpret_cast<float4*>(&redbuf[wave][k4]) = accPT;
    __syncthreads();
    if (tid < T) {
        float s = 0.f;
        #pragma unroll
        for (int w = 0; w < NW; ++w) s += redbuf[w][tid];
        atomAddF(oT + (b*T + i)*T + tid, s);     // pt[b,i,k]
    }
}

// ---------- host ----------
extern "C" void kernel_launch(void* const* d_in, const int* in_sizes, int n_in,
                              void* d_out, int out_size, void* d_ws, size_t ws_size,
                              hipStream_t stream) {
    const float* span = (const float*)d_in[0];
    const float* ph   = (const float*)d_in[1];
    const float* pt   = (const float*)d_in[2];
    const float* rpsh = (const float*)d_in[3];
    const float* rpst = (const float*)d_in[4];
    const float* fpsh = (const float*)d_in[5];
    const float* fpst = (const float*)d_in[6];
    const float* sh   = (const float*)d_in[7];
    const float* st   = (const float*)d_in[8];
    const float* ch   = (const float*)d_in[9];
    const float* ct   = (const float*)d_in[10];
    const float* m_r  = (const float*)d_in[11];   // mask2o_p2rspan
    const float* m_s  = (const float*)d_in[12];   // mask2o_psib
    const float* m_c  = (const float*)d_in[13];   // mask2o_pcop
    const float* m_f  = (const float*)d_in[14];   // mask2o_pspan2r

    const int N = B * T2;                 // 65536 per 3D tensor
    float* oS = (float*)d_out;
    float* oH = oS + N;
    float* oT = oH + N;

    float* qs = (float*)d_ws;
    float* qh = qs + N;
    float* qt = qh + N;

    const float* curS = span;
    const float* curH = ph;
    const float* curT = pt;

    for (int it = 0; it < 3; ++it) {
        k_sigmoid_init<<<N / 256, 256, 0, stream>>>(curS, curH, curT,
                                                    span, ph, pt,
                                                    qs, qh, qt,
                                                    oS, oH, oT);
        k_P<<<B * T, 256, 0, stream>>>(qs, qh, qt, rpsh, rpst, m_r, fpst, m_f,
                                       oS, oH, oT);
        k_Q<<<B * T, 256, 0, stream>>>(qs, qh, qt, fpsh, m_f, sh, st, m_s,
                                       ch, ct, m_c, rpst, m_r,
                                       oS, oH, oT);
        curS = oS; curH = oH; curT = oT;
    }
}